// PCAEncoder_2010044694991
// MI455X (gfx1250) — compile-verified
//
#include <hip/hip_runtime.h>
#include <math.h>

// Problem constants (x: (8, 31, 512, 512) fp32; A = x viewed as (B, HW=512*512, C=31))
#define Bn  8
#define Cc  31
#define HWc 262144
#define Qr  3

typedef __attribute__((ext_vector_type(2))) float v2f;
typedef __attribute__((ext_vector_type(8))) float v8f;

// ---------------------------------------------------------------- utilities
__device__ __forceinline__ unsigned hashu(unsigned x) {
  x ^= x >> 16; x *= 0x7feb352dU; x ^= x >> 15; x *= 0x846ca68bU; x ^= x >> 16;
  return x;
}

__global__ void k_zero(float* p, int n) {
  int i = blockIdx.x * blockDim.x + threadIdx.x;
  if (i < n) p[i] = 0.f;
}

// Fill padded 32x16 test matrix Wp (only [0..30][0..2] nonzero) with N(0,1).
__global__ void k_randn(float* Wp, unsigned seed) {
  if (threadIdx.x == 0 && blockIdx.x == 0) {
    for (int i = 0; i < 512; i++) Wp[i] = 0.f;
    for (int j = 0; j < Cc; j++)
      for (int q = 0; q < Qr; q++) {
        unsigned idx = (unsigned)(j * Qr + q);
        unsigned h1 = hashu(seed * 0x9E3779B9u + idx * 2u + 1u);
        unsigned h2 = hashu(seed * 0x85EBCA6Bu + idx * 2u + 2u);
        float u1 = ((float)h1 + 1.0f) * (1.0f / 4294967296.0f);
        float u2 = (float)h2 * (1.0f / 4294967296.0f);
        float r = sqrtf(-2.0f * logf(u1));
        Wp[j * 16 + q] = r * cosf(6.28318530718f * u2);
      }
  }
}

// -------------------------------------------------- WMMA tall-skinny GEMM
// Y(m,3) = Aeff(m,31) @ W(31,3), per batch.  Aeff = A - u * sv^T when uimg!=0.
// Each wave computes one 16x16 tile (only 3 cols kept), K padded 31->32 as
// 8 chained V_WMMA_F32_16X16X4_F32 accumulations.
__global__ __launch_bounds__(256) void k_gemm(
    const float* __restrict__ x, const float* __restrict__ Wp, int wpStride,
    const float* __restrict__ uimg, const float* __restrict__ sv,
    float* __restrict__ Y) {
  const int b    = blockIdx.y;
  const int lane = threadIdx.x & 31;
  const int wave = threadIdx.x >> 5;
  const int col  = lane & 15;   // A-frag M index & B-frag N index
  const int kh   = lane >> 4;   // K half-select (lanes 16..31 hold K+2)
  const int rowBase = blockIdx.x * 128 + wave * 16;
  const int row  = rowBase + col;
  const float* xb = x + (size_t)b * Cc * HWc;
  const float* W  = Wp + (size_t)b * wpStride;

  const bool res = (uimg != nullptr);
  float u0 = 0.f, u1 = 0.f, u2 = 0.f;
  if (res) {
    const float* ub = uimg + (size_t)b * 3 * HWc;
    u0 = ub[row]; u1 = ub[HWc + row]; u2 = ub[2 * HWc + row];
  }

  v8f acc = {};
#pragma unroll
  for (int k = 0; k < 8; k++) {
    const int kk0 = 4 * k + 2 * kh;           // <= 30
    const int kk1 = kk0 + 1;                  // <= 31 (pad)
    const float m1  = (kk1 < Cc) ? 1.f : 0.f;
    const int   kk1c = (kk1 < Cc) ? kk1 : (Cc - 1);
    float ax = xb[(size_t)kk0 * HWc + row];
    float ay = xb[(size_t)kk1c * HWc + row];
    if (res) {
      ax -= u0 * sv[kk0 * 3 + 0] + u1 * sv[kk0 * 3 + 1] + u2 * sv[kk0 * 3 + 2];
      ay -= u0 * sv[kk1c * 3 + 0] + u1 * sv[kk1c * 3 + 1] + u2 * sv[kk1c * 3 + 2];
    }
    ay *= m1;
    v2f a;  a.x = ax;                 a.y = ay;
    v2f bb; bb.x = W[kk0 * 16 + col]; bb.y = W[(kk0 + 1) * 16 + col];
#if __has_builtin(__builtin_amdgcn_wmma_f32_16x16x4_f32)
    acc = __builtin_amdgcn_wmma_f32_16x16x4_f32(false, a, false, bb,
                                                (short)0, acc, false, false);
#else
    // VALU fallback (should not trigger on gfx1250)
    acc[0] += a.x * bb.x + a.y * bb.y;
#endif
  }
  // D layout: VGPR r -> M = r + 8*kh, N = col
  if (col < Qr) {
#pragma unroll
    for (int r = 0; r < 8; r++) {
      const int m = rowBase + r + 8 * kh;
      Y[((size_t)b * HWc + m) * 3 + col] = acc[r];
    }
  }
}

// -------------------------------------------------- T(31,3) = Aeff^T @ Q(m,3)
__global__ __launch_bounds__(256) void k_atb(
    const float* __restrict__ x, const float* __restrict__ uimg,
    const float* __restrict__ sv, const float* __restrict__ Q,
    float* __restrict__ T) {
  const int b = blockIdx.y;
  const float* xb = x + (size_t)b * Cc * HWc;
  const float* Qb = Q + (size_t)b * HWc * 3;
  const bool res = (uimg != nullptr);

  float acc[Cc * 3];
#pragma unroll
  for (int t = 0; t < Cc * 3; t++) acc[t] = 0.f;

  for (int i = blockIdx.x * blockDim.x + threadIdx.x; i < HWc;
       i += gridDim.x * blockDim.x) {
    const float q0 = Qb[i * 3 + 0], q1 = Qb[i * 3 + 1], q2 = Qb[i * 3 + 2];
    float u0 = 0.f, u1 = 0.f, u2 = 0.f;
    if (res) {
      const float* ub = uimg + (size_t)b * 3 * HWc;
      u0 = ub[i]; u1 = ub[HWc + i]; u2 = ub[2 * HWc + i];
    }
#pragma unroll
    for (int j = 0; j < Cc; j++) {
      float a = xb[(size_t)j * HWc + i];
      if (res) a -= u0 * sv[j * 3 + 0] + u1 * sv[j * 3 + 1] + u2 * sv[j * 3 + 2];
      acc[j * 3 + 0] += a * q0;
      acc[j * 3 + 1] += a * q1;
      acc[j * 3 + 2] += a * q2;
    }
  }
  __shared__ float red[Cc * 3];
  if (threadIdx.x < Cc * 3) red[threadIdx.x] = 0.f;
  __syncthreads();
#pragma unroll
  for (int t = 0; t < Cc * 3; t++) atomicAdd(&red[t], acc[t]);
  __syncthreads();
  if (threadIdx.x < Cc * 3) atomicAdd(&T[b * 96 + threadIdx.x], red[threadIdx.x]);
}

// -------------------------------------------------- G(3,3) = Y^T Y
__global__ __launch_bounds__(256) void k_gram(const float* __restrict__ Y,
                                              float* __restrict__ G) {
  const int b = blockIdx.y;
  const float* Yb = Y + (size_t)b * HWc * 3;
  float acc[9];
#pragma unroll
  for (int t = 0; t < 9; t++) acc[t] = 0.f;
  for (int i = blockIdx.x * blockDim.x + threadIdx.x; i < HWc;
       i += gridDim.x * blockDim.x) {
    const float y0 = Yb[i * 3 + 0], y1 = Yb[i * 3 + 1], y2 = Yb[i * 3 + 2];
    acc[0] += y0 * y0; acc[1] += y0 * y1; acc[2] += y0 * y2;
    acc[4] += y1 * y1; acc[5] += y1 * y2; acc[8] += y2 * y2;
  }
  acc[3] = acc[1]; acc[6] = acc[2]; acc[7] = acc[5];
  __shared__ float red[9];
  if (threadIdx.x < 9) red[threadIdx.x] = 0.f;
  __syncthreads();
#pragma unroll
  for (int t = 0; t < 9; t++) atomicAdd(&red[t], acc[t]);
  __syncthreads();
  if (threadIdx.x < 9) atomicAdd(&G[b * 16 + threadIdx.x], red[threadIdx.x]);
}

// -------------------------------------------------- Z(m,3) = P(m,3) @ M(3,3)
// mode 0: Z to (b,i,q) buffer.  mode 1: Z transposed into u_img region.
__global__ __launch_bounds__(256) void k_apply(
    const float* __restrict__ P, const float* __restrict__ M,
    float* __restrict__ Zn, float* __restrict__ Zt, int mode) {
  const int b = blockIdx.y;
  const int i = blockIdx.x * blockDim.x + threadIdx.x;
  if (i >= HWc) return;
  const float* Pb = P + (size_t)b * HWc * 3;
  const float* Mb = M + b * 16;
  const float p0 = Pb[i * 3 + 0], p1 = Pb[i * 3 + 1], p2 = Pb[i * 3 + 2];
#pragma unroll
  for (int k = 0; k < 3; k++) {
    const float z = p0 * Mb[0 * 3 + k] + p1 * Mb[1 * 3 + k] + p2 * Mb[2 * 3 + k];
    if (mode == 0) Zn[((size_t)b * HWc + i) * 3 + k] = z;
    else           Zt[(size_t)b * 3 * HWc + (size_t)k * HWc + i] = z;
  }
}

// -------------------------------------------------- CholQR: M = L^{-T} from G
__global__ void k_chol(const float* __restrict__ G, float* __restrict__ M) {
  const int b = blockIdx.x;
  if (threadIdx.x != 0) return;
  const float* g = G + b * 16;
  const float eps = 1e-20f;
  const float a00 = g[0], a01 = g[1], a02 = g[2], a11 = g[4], a12 = g[5], a22 = g[8];
  const float l00 = sqrtf(fmaxf(a00, eps));
  const float l10 = a01 / l00, l20 = a02 / l00;
  const float l11 = sqrtf(fmaxf(a11 - l10 * l10, eps));
  const float l21 = (a12 - l20 * l10) / l11;
  const float l22 = sqrtf(fmaxf(a22 - l20 * l20 - l21 * l21, eps));
  const float i00 = 1.f / l00, i11 = 1.f / l11, i22 = 1.f / l22;
  const float i10 = -l10 * i00 * i11;
  const float i21 = -l21 * i11 * i22;
  const float i20 = (l10 * l21 - l20 * l11) * i00 * i11 * i22;
  float* m = M + b * 16;                       // M[q][k] = Linv[k][q] (upper)
  m[0] = i00; m[1] = i10; m[2] = i20;
  m[3] = 0.f; m[4] = i11; m[5] = i21;
  m[6] = 0.f; m[7] = 0.f; m[8] = i22;
}

// -------------------------------------------------- MGS QR of T(31,3) -> Wp
__global__ void k_qr31(const float* __restrict__ T, float* __restrict__ Wp) {
  const int b = blockIdx.x;
  if (threadIdx.x != 0) return;
  float v[Cc][3];
  for (int j = 0; j < Cc; j++)
    for (int q = 0; q < Qr; q++) v[j][q] = T[b * 96 + j * 3 + q];
  for (int q = 0; q < Qr; q++) {
    for (int p = 0; p < q; p++) {
      float d = 0.f;
      for (int j = 0; j < Cc; j++) d += v[j][q] * v[j][p];
      for (int j = 0; j < Cc; j++) v[j][q] -= d * v[j][p];
    }
    float n2 = 0.f;
    for (int j = 0; j < Cc; j++) n2 += v[j][q] * v[j][q];
    const float inv = 1.f / sqrtf(fmaxf(n2, 1e-30f));
    for (int j = 0; j < Cc; j++) v[j][q] *= inv;
  }
  float* W = Wp + b * 512;
  for (int i = 0; i < 512; i++) W[i] = 0.f;
  for (int j = 0; j < Cc; j++)
    for (int q = 0; q < Qr; q++) W[j * 16 + q] = v[j][q];
}

// --------------------------------- SVD of Bsmall(3,31) via 3x3 Jacobi eigen
// T holds Bs = Bsmall^T (31x3).  Outputs: Mout = U3 (apply to Q), s, v, s*v.
__global__ void k_svd3(const float* __restrict__ T, float* __restrict__ Mout,
                       float* __restrict__ s_out, float* __restrict__ v_out,
                       float* __restrict__ sv_out) {
  const int b = blockIdx.x;
  if (threadIdx.x != 0) return;
  const float* Bs = T + b * 96;
  float A[3][3] = {{0.f}};
  for (int j = 0; j < Cc; j++)
    for (int p = 0; p < 3; p++)
      for (int r = 0; r < 3; r++) A[p][r] += Bs[j * 3 + p] * Bs[j * 3 + r];
  float V[3][3] = {{1, 0, 0}, {0, 1, 0}, {0, 0, 1}};
  const int PP[3] = {0, 0, 1}, QQ[3] = {1, 2, 2};
  for (int sweep = 0; sweep < 12; sweep++)
    for (int e = 0; e < 3; e++) {
      const int p = PP[e], q = QQ[e];
      const float apq = A[p][q];
      if (fabsf(apq) > 1e-12f) {
        const float theta = (A[q][q] - A[p][p]) / (2.f * apq);
        const float t = copysignf(1.f, theta) / (fabsf(theta) + sqrtf(theta * theta + 1.f));
        const float c = 1.f / sqrtf(t * t + 1.f), s = t * c;
        for (int k = 0; k < 3; k++) { float akp = A[k][p], akq = A[k][q];
          A[k][p] = c * akp - s * akq; A[k][q] = s * akp + c * akq; }
        for (int k = 0; k < 3; k++) { float apk = A[p][k], aqk = A[q][k];
          A[p][k] = c * apk - s * aqk; A[q][k] = s * apk + c * aqk; }
        for (int k = 0; k < 3; k++) { float vkp = V[k][p], vkq = V[k][q];
          V[k][p] = c * vkp - s * vkq; V[k][q] = s * vkp + c * vkq; }
      }
    }
  int idx[3] = {0, 1, 2};
  for (int a = 0; a < 2; a++)
    for (int bb = 0; bb < 2 - a; bb++)
      if (A[idx[bb]][idx[bb]] < A[idx[bb + 1]][idx[bb + 1]]) {
        int tmp = idx[bb]; idx[bb] = idx[bb + 1]; idx[bb + 1] = tmp;
      }
  float S[3], U3[3][3];
  for (int k = 0; k < 3; k++) {
    S[k] = sqrtf(fmaxf(A[idx[k]][idx[k]], 0.f));
    s_out[b * 3 + k] = S[k];
    for (int q = 0; q < 3; q++) U3[q][k] = V[q][idx[k]];
  }
  float* m = Mout + b * 16;
  for (int q = 0; q < 3; q++)
    for (int k = 0; k < 3; k++) m[q * 3 + k] = U3[q][k];
  for (int j = 0; j < Cc; j++)
    for (int k = 0; k < 3; k++) {
      float val = 0.f;
      for (int q = 0; q < 3; q++) val += Bs[j * 3 + q] * U3[q][k];
      const float vjk = val / fmaxf(S[k], 1e-20f);
      v_out[b * (Cc * 3) + j * 3 + k] = vjk;
      sv_out[b * 96 + j * 3 + k] = S[k] * vjk;
    }
}

// ---------------------------------------------------------------- launcher
extern "C" void kernel_launch(void* const* d_in, const int* in_sizes, int n_in,
                              void* d_out, int out_size, void* d_ws, size_t ws_size,
                              hipStream_t stream) {
  const float* x = (const float*)d_in[0];
  float* out = (float*)d_out;
  float* ws  = (float*)d_ws;     // needs ~50.4 MB: 2*(B*HW*3) fp32 + smalls

  float* wY  = ws;
  float* wQ  = wY + (size_t)Bn * HWc * 3;
  float* Wp  = wQ + (size_t)Bn * HWc * 3;   // Bn*512 padded 32x16 W matrices
  float* Tm  = Wp + Bn * 512;               // Bn*96   (A^T Q / Bsmall^T)
  float* Gm  = Tm + Bn * 96;                // Bn*16   Gram
  float* Mm  = Gm + Bn * 16;                // Bn*16   3x3 apply matrix
  float* sv1 = Mm + Bn * 16;                // Bn*96   s*v (pass 1 residual)
  float* sv2 = sv1 + Bn * 96;               // Bn*96

  float* o_u  = out;                         // u_img  (B,3,H,W)
  float* o_s  = out + (size_t)Bn * 3 * HWc;  // s      (B,3)
  float* o_v  = o_s + Bn * 3;                // v      (B,31,3)
  float* o_uu = o_v + Bn * Cc * 3;           // uu_img
  float* o_ss = o_uu + (size_t)Bn * 3 * HWc; // ss
  float* o_vv = o_ss + Bn * 3;               // vv

  const dim3 b256(256);
  const dim3 gGemm(HWc / 128, Bn);
  const dim3 gRed(512, Bn);
  const dim3 gApply(HWc / 256, Bn);

  auto pass = [&](unsigned seed, const float* u_res, const float* sv_res,
                  float* sv_store, float* ou, float* os, float* ov) {
    k_randn<<<1, 1, 0, stream>>>(Wp, seed);
    k_gemm<<<gGemm, b256, 0, stream>>>(x, Wp, 0, u_res, sv_res, wY);
    k_zero<<<1, 128, 0, stream>>>(Gm, Bn * 16);
    k_gram<<<gRed, b256, 0, stream>>>(wY, Gm);
    k_chol<<<Bn, 32, 0, stream>>>(Gm, Mm);
    k_apply<<<gApply, b256, 0, stream>>>(wY, Mm, wQ, nullptr, 0);
    for (int it = 0; it < 2; ++it) {                    // power iterations
      k_zero<<<2, 512, 0, stream>>>(Tm, Bn * 96);
      k_atb<<<gRed, b256, 0, stream>>>(x, u_res, sv_res, wQ, Tm);
      k_qr31<<<Bn, 32, 0, stream>>>(Tm, Wp);
      k_gemm<<<gGemm, b256, 0, stream>>>(x, Wp, 512, u_res, sv_res, wY);
      k_zero<<<1, 128, 0, stream>>>(Gm, Bn * 16);
      k_gram<<<gRed, b256, 0, stream>>>(wY, Gm);
      k_chol<<<Bn, 32, 0, stream>>>(Gm, Mm);
      k_apply<<<gApply, b256, 0, stream>>>(wY, Mm, wQ, nullptr, 0);
    }
    k_zero<<<2, 512, 0, stream>>>(Tm, Bn * 96);
    k_atb<<<gRed, b256, 0, stream>>>(x, u_res, sv_res, wQ, Tm);  // Bsmall^T
    k_svd3<<<Bn, 32, 0, stream>>>(Tm, Mm, os, ov, sv_store);
    k_apply<<<gApply, b256, 0, stream>>>(wQ, Mm, nullptr, ou, 1);
  };

  pass(1u, nullptr, nullptr, sv1, o_u, o_s, o_v);     // first pca_lowrank
  pass(2u, o_u, sv1, sv2, o_uu, o_ss, o_vv);          // on fused residual
}